// TrainedDOAspireConSent_65695819759800
// MI455X (gfx1250) — compile-verified
//
#include <hip/hip_runtime.h>

#define B_   32
#define S_   512
#define D_   1536
#define MAXS 24

typedef float v2f __attribute__((ext_vector_type(2)));
typedef float v8f __attribute__((ext_vector_type(8)));

// grid: (D_/128, B_) = (12, 32); block: 256 threads = 8 waves (wave32).
// Each wave owns a 16-column slice of D; each block owns a 128-column slice
// of one batch. hidden is read exactly once across the grid (HBM-bound,
// ~100 MB => ~4.5 us floor at 23.3 TB/s).
__global__ __launch_bounds__(256)
void sentpool_wmma_kernel(const float* __restrict__ hidden,
                          const int*   __restrict__ attn_mask,
                          const int*   __restrict__ sent_ids,
                          float*       __restrict__ out_doc,   // [B, D]
                          float*       __restrict__ out_sent)  // [B, MAXS, D]
{
    __shared__ int   s_sid[S_];
    __shared__ float s_cnt[32];
    __shared__ float s_inv[32];
    __shared__ int   s_asum;
    __shared__ int   s_lp;

    const int b        = blockIdx.y;
    const int colBlock = blockIdx.x * 128;
    const int tid      = threadIdx.x;
    const int lane     = tid & 31;
    const int wave     = tid >> 5;
    const int half     = (lane >> 4) & 1;   // which half of the wave
    const int n        = lane & 15;         // column-in-tile AND seg-row-in-tile

    if (tid < 32) s_cnt[tid] = 0.0f;
    if (tid == 0) { s_asum = 0; s_lp = 0; }
    __syncthreads();

    // ---- stage sent_ids into LDS, build segment counts + attn-mask sum ----
    int asum = 0;
    for (int i = tid; i < S_; i += 256) {
        int sid = sent_ids[b * S_ + i];
        s_sid[i] = sid;
        atomicAdd(&s_cnt[sid & 31], 1.0f);   // ds_add_f32
        asum += attn_mask[b * S_ + i];
    }
    atomicAdd(&s_asum, asum);
    if (tid < B_) atomicAdd(&s_lp, attn_mask[tid * S_ + (S_ - 1)]);
    __syncthreads();

    // per-segment reciprocal of clamped count (32 exact divides per block;
    // epilogue then needs only one v_mul per output element)
    if (tid < 32) {
        float c = s_cnt[tid];
        s_inv[tid] = 1.0f / ((c < 1.0f) ? 1.0f : c);
    }

    // ---- last_token_pool row selection (uniform per block) ----
    int row = (s_lp == B_) ? (S_ - 1) : (s_asum - 1);
    if (row < 0) row = 0;

    const float* hb = hidden + (size_t)b * S_ * D_;

    // doc_reps for this block's 128-column slice
    if (tid < 128) {
        int c = colBlock + tid;
        out_doc[(size_t)b * D_ + c] = hb[(size_t)row * D_ + c];
    }

    // ---- segment sums via V_WMMA_F32_16X16X4_F32 ----
    // A (16x4 f32): one-hot(seg==sent_id); lane L holds K = 2*half, 2*half+1
    // B (4x16 f32): hidden tile;           lane L holds K = 2*half, 2*half+1
    // D (16x16 f32): [seg, col]; vgpr r, lane L -> seg = r + 8*half, col = n
    const int col = colBlock + wave * 16 + n;

    v8f acc0 = {};   // segments 0..15
    v8f acc1 = {};   // segments 16..31 (25..31 never matched -> zero)

    #pragma unroll 8
    for (int t0 = 0; t0 < S_; t0 += 4) {
        const int tA = t0 + 2 * half;
        const int sA = s_sid[tA];
        const int sB = s_sid[tA + 1];

        v2f a0, a1, bb;
        a0.x = (sA == n)        ? 1.0f : 0.0f;
        a0.y = (sB == n)        ? 1.0f : 0.0f;
        a1.x = (sA == n + 16)   ? 1.0f : 0.0f;
        a1.y = (sB == n + 16)   ? 1.0f : 0.0f;
        bb.x = hb[(size_t)tA * D_ + col];
        bb.y = hb[(size_t)(tA + 1) * D_ + col];

        acc0 = __builtin_amdgcn_wmma_f32_16x16x4_f32(
                   false, a0, false, bb, (short)0, acc0, false, false);
        acc1 = __builtin_amdgcn_wmma_f32_16x16x4_f32(
                   false, a1, false, bb, (short)0, acc1, false, false);
    }

    __syncthreads();   // s_inv ready (also keeps s_inv write ordered)

    // ---- scale by reciprocal counts, store sent_reps ----
    #pragma unroll
    for (int r = 0; r < 8; ++r) {
        const int seg0 = r + 8 * half;          // 0..15, always valid
        out_sent[((size_t)b * MAXS + seg0) * D_ + col] = acc0[r] * s_inv[seg0];

        const int seg1 = 16 + r + 8 * half;     // 16..31; valid only < 24
        if (seg1 < MAXS) {
            out_sent[((size_t)b * MAXS + seg1) * D_ + col] = acc1[r] * s_inv[seg1];
        }
    }
}

extern "C" void kernel_launch(void* const* d_in, const int* in_sizes, int n_in,
                              void* d_out, int out_size, void* d_ws, size_t ws_size,
                              hipStream_t stream) {
    (void)in_sizes; (void)n_in; (void)d_ws; (void)ws_size; (void)out_size;

    const float* hidden    = (const float*)d_in[0];
    const int*   attn_mask = (const int*)  d_in[1];
    const int*   sent_ids  = (const int*)  d_in[2];

    float* out_doc  = (float*)d_out;                      // [B, D]
    float* out_sent = (float*)d_out + (size_t)B_ * D_;    // [B, MAXS, D]

    dim3 grid(D_ / 128, B_);
    dim3 block(256);
    sentpool_wmma_kernel<<<grid, block, 0, stream>>>(
        hidden, attn_mask, sent_ids, out_doc, out_sent);
}